// llama_kvcache_50869592655594
// MI455X (gfx1250) — compile-verified
//
#include <hip/hip_runtime.h>
#include <stdint.h>

// KV-cache prefill insert: out[2, B, S_MAX, H, D] fp32.
//   out[kv][b][s] = (start_pos <= s < start_pos+S_NEW) ? key/value[b][s-start_pos]
//                                                      : cache_key/value[b][s]
// Pure streaming copy (~640 MB traffic, 0 FLOPs) -> HBM-bound (~27 us @ 23.3 TB/s).
// CDNA5 async global<->LDS data-mover path (ASYNCcnt) + non-temporal hints.
// One tile == one (kv,b,s) row == 256 float4 == 4 KB, so the source-region
// decision is per-tile and all 8 b128 async ops share one address pair with
// immediate offsets (INST_OFFSET applies to BOTH global and LDS addresses).

#define S_NEW   4096
#define S_MAX   8192

#define ROWS_TOTAL  65536u              // 2 * B * S_MAX = 2*4*8192
#define ROW_V4      256u                // H*D/4 float4 per row
#define ASYNC_U     8                   // 8 x (32 lanes x 16B) = 4 KB per row
#define WAVE_BUF    (ASYNC_U * 32 * 16) // 4 KB per LDS buffer per wave
#define WAVES_PB    8                   // 256 threads / 32

template <int OFF>
__device__ __forceinline__ void async_g2l(uint32_t lds, const void* g) {
    asm volatile("global_load_async_to_lds_b128 %0, %1, off offset:%2 th:TH_LOAD_NT"
                 :: "v"(lds), "v"(g), "n"(OFF) : "memory");
}

template <int OFF>
__device__ __forceinline__ void async_l2g(void* g, uint32_t lds) {
    asm volatile("global_store_async_from_lds_b128 %0, %1, off offset:%2 th:TH_STORE_NT"
                 :: "v"(g), "v"(lds), "n"(OFF) : "memory");
}

__global__ __launch_bounds__(256) void kv_prefill_async_copy(
    const float4* __restrict__ key,
    const float4* __restrict__ value,
    const float4* __restrict__ cache_k,
    const float4* __restrict__ cache_v,
    const int*    __restrict__ start_pos,
    float4*       __restrict__ out)
{
    // per-wave double-buffered staging: 8 waves * 2 bufs * 4KB = 64KB
    __shared__ char smem[WAVES_PB * 2 * WAVE_BUF];

    const unsigned lane = threadIdx.x & 31u;
    const unsigned wave = threadIdx.x >> 5;

    const int sp = *start_pos;   // device scalar (uniform)

    const uint32_t ldsWaveBase =
        (uint32_t)(uintptr_t)(&smem[wave * (2 * WAVE_BUF)]) + lane * 16u;

    const unsigned rstride = gridDim.x * WAVES_PB;
    unsigned row = blockIdx.x * WAVES_PB + wave;
    unsigned it  = 0;                                  // parity for double-buffer

    for (; row < ROWS_TOTAL; row += rstride, ++it) {
        const uint32_t lds = ldsWaveBase + (it & 1u) * WAVE_BUF;

        // row -> (kv, b, s); rows-per-kv = B*S_MAX = 2^15, rows-per-batch = 2^13
        const unsigned kv = row >> 15;
        const unsigned b  = (row >> 13) & 3u;
        const unsigned s  = row & (S_MAX - 1u);
        const int    srel = (int)s - sp;

        const float4* srcp;
        if ((unsigned)srel < (unsigned)S_NEW) {
            const float4* base = kv ? value : key;
            srcp = base + ((((size_t)b * S_NEW + (unsigned)srel) << 8));
        } else {
            const float4* base = kv ? cache_v : cache_k;
            srcp = base + ((size_t)(row & 0x7FFFu) << 8);  // cache layout == out layout
        }
        srcp += lane;
        float4* dstp = out + (((size_t)row << 8) + lane);

        // ---- 8 async b128 loads (global -> LDS), shared addr, imm offsets ----
        async_g2l<0 * 512>(lds, srcp);
        async_g2l<1 * 512>(lds, srcp);
        async_g2l<2 * 512>(lds, srcp);
        async_g2l<3 * 512>(lds, srcp);
        async_g2l<4 * 512>(lds, srcp);
        async_g2l<5 * 512>(lds, srcp);
        async_g2l<6 * 512>(lds, srcp);
        async_g2l<7 * 512>(lds, srcp);

        // loads must land in LDS before the dependent stores read it
        asm volatile("s_wait_asynccnt 0" ::: "memory");

        // ---- 8 async b128 stores (LDS -> global) ----
        async_l2g<0 * 512>(dstp, lds);
        async_l2g<1 * 512>(dstp, lds);
        async_l2g<2 * 512>(dstp, lds);
        async_l2g<3 * 512>(dstp, lds);
        async_l2g<4 * 512>(dstp, lds);
        async_l2g<5 * 512>(dstp, lds);
        async_l2g<6 * 512>(dstp, lds);
        async_l2g<7 * 512>(dstp, lds);
        // stores drain while the next row loads into the other buffer; the next
        // iteration's s_wait_asynccnt 0 (and s_endpgm's implicit wait-idle)
        // close the buffer-reuse hazard.
    }
}

extern "C" void kernel_launch(void* const* d_in, const int* in_sizes, int n_in,
                              void* d_out, int out_size, void* d_ws, size_t ws_size,
                              hipStream_t stream) {
    const float4* key     = (const float4*)d_in[0];
    const float4* value   = (const float4*)d_in[1];
    const float4* cache_k = (const float4*)d_in[2];
    const float4* cache_v = (const float4*)d_in[3];
    const int*    sp      = (const int*)d_in[4];
    float4*       out     = (float4*)d_out;

    dim3 grid(2048), block(256);   // 16K waves, 4 rows each
    hipLaunchKernelGGL(kv_prefill_async_copy, grid, block, 0, stream,
                       key, value, cache_k, cache_v, sp, out);
}